// SpixelAggr_avr_14499809591944
// MI455X (gfx1250) — compile-verified
//
#include <hip/hip_runtime.h>
#include <hip/hip_bf16.h>

#define N_PIX 1048576
#define C_CH  64
#define K_SEG 2048

typedef __attribute__((ext_vector_type(2))) float v2f;
typedef __attribute__((ext_vector_type(8))) float v8f;

// ---------------------------------------------------------------------------
// Pass 0: zero sums (d_out, K*C floats) and counts (workspace, K floats)
// ---------------------------------------------------------------------------
__global__ void spix_zero_kernel(float* __restrict__ sums, float* __restrict__ counts) {
    int i = blockIdx.x * blockDim.x + threadIdx.x;
    if (i < K_SEG * C_CH) sums[i] = 0.0f;
    if (i < K_SEG)        counts[i] = 0.0f;
}

// ---------------------------------------------------------------------------
// Pass 1: gather + scatter-add.  One wave32 per pixel; 32 lanes cover the
// 64-channel row as float2 (global_load_b64, 256B contiguous per row).
// Sums array is 512KB -> resident in L2, so f32 atomics stay on-chip.
// ---------------------------------------------------------------------------
__global__ void spix_aggregate_kernel(const float* __restrict__ in,
                                      const long long* __restrict__ seg,
                                      const long long* __restrict__ coor,
                                      float* __restrict__ sums,
                                      float* __restrict__ counts) {
    int pix  = (blockIdx.x * blockDim.x + threadIdx.x) >> 5;  // wave id = pixel
    int lane = threadIdx.x & 31;
    if (pix >= N_PIX) return;

    long long src = coor[pix];   // wave-uniform
    long long lab = seg[pix];    // wave-uniform

    const float2* row = reinterpret_cast<const float2*>(in + src * (long long)C_CH);
    float2 v = row[lane];        // lanes 0..31 -> channels 0..63, coalesced 256B

    float* dst = sums + lab * (long long)C_CH + lane * 2;
    atomicAdd(dst + 0, v.x);
    atomicAdd(dst + 1, v.y);
    if (lane == 0) atomicAdd(counts + lab, 1.0f);
}

// ---------------------------------------------------------------------------
// Pass 2: mean = diag(1/count) @ sums, tile-wise via V_WMMA_F32_16X16X4_F32.
// A 16x16x16 row-scale matmul = 4 chained 16x16x4 WMMAs with C accumulation:
//   D = sum_j A_j(16x4) @ B_j(4x16),  A_j[m][k] = inv[m] iff m == 4j+k.
// One wave per (16 labels x 16 channels) tile; 512 tiles total.
//
// Layouts per ISA 7.12.2 (wave32):
//   A 16x4 f32 : lanes 0-15 M=0..15 hold K=0 (v0) / K=1 (v1);
//                lanes 16-31 hold K=2 (v0) / K=3 (v1).
//   B 4x16     : mirrored (rows striped across lanes within a VGPR).
//   C/D 16x16  : VGPR r -> M=r (lanes 0-15, N=lane), M=r+8 (lanes 16-31).
// ---------------------------------------------------------------------------
__global__ void spix_normalize_kernel(float* __restrict__ sums,          // in/out (d_out)
                                      const float* __restrict__ counts) {
    int tile = blockIdx.x * (blockDim.x >> 5) + (threadIdx.x >> 5);
    int lane = threadIdx.x & 31;

    int kb = (tile >> 2) * 16;   // label-group base   (128 groups)
    int cb = (tile & 3) * 16;    // channel-group base (4 groups)

    int m  = lane & 15;
    int k0 = (lane < 16) ? 0 : 2;   // which K pair this half-wave holds

    float cnt = counts[kb + m];
    float inv = 1.0f / fmaxf(cnt, 1.0f);

    v8f acc = {};
#pragma unroll
    for (int j = 0; j < 4; ++j) {
        v2f a, b;
        // A_j: diagonal slice scaled by inv[m]
        a.x = (m == j * 4 + k0)     ? inv : 0.0f;
        a.y = (m == j * 4 + k0 + 1) ? inv : 0.0f;
        // B_j: rows 4j+k0, 4j+k0+1 of the sums tile, column cb+m
        b.x = sums[(kb + j * 4 + k0)     * C_CH + cb + m];
        b.y = sums[(kb + j * 4 + k0 + 1) * C_CH + cb + m];
        // 8-arg form: (neg_a, A, neg_b, B, c_mod, C, reuse_a, reuse_b)
        acc = __builtin_amdgcn_wmma_f32_16x16x4_f32(false, a, false, b,
                                                    (short)0, acc, false, false);
    }

    // Write D tile back in place (all reads above completed within this wave).
#pragma unroll
    for (int r = 0; r < 8; ++r) {
        int mm = (lane < 16) ? r : (r + 8);
        sums[(kb + mm) * C_CH + cb + m] = acc[r];
    }
}

// ---------------------------------------------------------------------------
extern "C" void kernel_launch(void* const* d_in, const int* in_sizes, int n_in,
                              void* d_out, int out_size, void* d_ws, size_t ws_size,
                              hipStream_t stream) {
    const float*     input = (const float*)d_in[0];
    const long long* seg   = (const long long*)d_in[1];
    const long long* coor  = (const long long*)d_in[2];
    float* sums   = (float*)d_out;      // [K, C]
    float* counts = (float*)d_ws;       // [K]

    // Pass 0: zero accumulators (harness poisons d_out/d_ws with 0xAA).
    spix_zero_kernel<<<(K_SEG * C_CH + 255) / 256, 256, 0, stream>>>(sums, counts);

    // Pass 1: one wave32 per pixel -> 8 pixels per 256-thread block.
    spix_aggregate_kernel<<<N_PIX / 8, 256, 0, stream>>>(input, seg, coor, sums, counts);

    // Pass 2: 512 tiles, one wave each -> 8 tiles per 256-thread block.
    spix_normalize_kernel<<<(K_SEG / 16) * (C_CH / 16) / 8, 256, 0, stream>>>(sums, counts);
}